// DetectionLoss_89618787599033
// MI455X (gfx1250) — compile-verified
//
#include <hip/hip_runtime.h>

// ---------------- problem constants ----------------
#define B_      32
#define P_      8192
#define G_      150
#define GPAD_   160
#define C_      8
#define PBLOCKS_ 16            // cost-kernel blocks per image (512 preds each)
#define THRESH_ENC 0xBFC00000u // order-preserving encoding of 1.5f

typedef __attribute__((ext_vector_type(2))) float v2f;
typedef __attribute__((ext_vector_type(8))) float v8f;
typedef __attribute__((ext_vector_type(4))) unsigned int u32x4;
typedef __attribute__((ext_vector_type(8))) int i32x8;
typedef __attribute__((ext_vector_type(4))) int i32x4;
typedef unsigned long long u64;
typedef unsigned int u32;

#if __has_builtin(__builtin_amdgcn_tensor_load_to_lds) && \
    __has_builtin(__builtin_amdgcn_s_wait_tensorcnt)
#define USE_TDM 1
#endif

// order-preserving float -> u32 (monotone for all finite floats)
__device__ __forceinline__ u32 encf(float f) {
    u32 u = __float_as_uint(f);
    return (u & 0x80000000u) ? ~u : (u | 0x80000000u);
}

__device__ __forceinline__ float giou4(float4 a, float4 b) {
    const float EPS = 1e-6f;
    float A1 = (a.z - a.x) * (a.w - a.y);
    float A2 = (b.z - b.x) * (b.w - b.y);
    float iw = fmaxf(fminf(a.z, b.z) - fmaxf(a.x, b.x), 0.f);
    float ih = fmaxf(fminf(a.w, b.w) - fmaxf(a.y, b.y), 0.f);
    float inter = iw * ih;
    float uni   = A1 + A2 - inter;
    float iou   = inter / (uni + EPS);
    float en = (fmaxf(a.z, b.z) - fminf(a.x, b.x)) * (fmaxf(a.w, b.w) - fminf(a.y, b.y));
    return iou - (en - uni) / (en + EPS);
}

__device__ __forceinline__ u64 shflxor64(u64 v, int m) {
    u32 lo = (u32)v, hi = (u32)(v >> 32);
    lo = __shfl_xor(lo, m, 32);
    hi = __shfl_xor(hi, m, 32);
    return ((u64)hi << 32) | lo;
}

// ---------------- kernel 0: workspace init ----------------
__global__ void det_init_ws(u64* slots, int nslots, float* accum, int naccum) {
    int i = blockIdx.x * blockDim.x + threadIdx.x;
    if (i < nslots) slots[i] = ~0ull;
    if (i < naccum) accum[i] = 0.f;
}

// ---------------- kernel 1: cost matrix via WMMA + per-gt packed min ----------------
// grid = B_*PBLOCKS_ blocks of 256 (8 wave32). Each block: 512 preds x 160 gts.
// gt boxes staged LDS-side by the Tensor Data Mover (1D 600-dword tile),
// per 16x16 tile: 3x v_wmma_f32_16x16x4_f32 (d2 bilinear form + two-chunk class matmul),
// GIoU elementwise on the 8 accumulator slots per lane, packed ds_min_u64 reduction.
__global__ __launch_bounds__(256) void det_cost_min(
    const float* __restrict__ pbox, const float* __restrict__ plog,
    const float* __restrict__ gbox, const int* __restrict__ glbl,
    u64* __restrict__ gmin)
{
    __shared__ float4 s_gbox[GPAD_];
    __shared__ float  s_gcx[GPAD_], s_gcy[GPAD_], s_gn[GPAD_];
    __shared__ int    s_glbl[GPAD_];
    __shared__ u64    s_gmin[GPAD_];

    const int bid  = blockIdx.x;
    const int b    = bid / PBLOCKS_;
    const int pblk = bid % PBLOCKS_;
    const int tid  = threadIdx.x;

#ifdef USE_TDM
    // ---- TDM staging: 1D tile of G_*4 = 600 dwords  global -> LDS ----
    if (tid < 32) {                       // wave 0 only (uniform branch)
        unsigned lds_off = (unsigned)(unsigned long long)&s_gbox[0];
        unsigned long long ga = (unsigned long long)(size_t)(gbox + (size_t)b * G_ * 4);
        u32x4 g0;
        g0.x = 1u;                                   // count=1 (user descriptor)
        g0.y = lds_off;                              // lds_addr
        g0.z = (u32)ga;                              // global_addr[31:0]
        g0.w = (u32)(ga >> 32) | (2u << 30);         // global_addr[56:32] | type=2
        i32x8 g1;
        g1[0] = 0x20000;                             // data_size=2 (4 bytes)
        g1[1] = (int)((600u & 0xFFFFu) << 16);       // tensor_dim0 lo16
        g1[2] = (int)(1u << 16);                     // tensor_dim0 hi16=0 | tensor_dim1 lo16=1
        g1[3] = (int)(600u << 16);                   // tensor_dim1 hi16=0 | tile_dim0=600
        g1[4] = 1;                                   // tile_dim1=1, tile_dim2=0
        g1[5] = 600;                                 // tensor_dim0_stride lo32
        g1[6] = 0;
        g1[7] = 0;
        i32x4 zz = {0, 0, 0, 0};                     // groups 2/3 unused (<=2D)
        i32x8 z8 = {0, 0, 0, 0, 0, 0, 0, 0};         // extra group (clang-23 6-arg form)
        __builtin_amdgcn_tensor_load_to_lds(g0, g1, zz, zz, z8, 0);
        __builtin_amdgcn_s_wait_tensorcnt((short)0); // TENSORcnt -> 0 before barrier
    }
    for (int t = tid; t < GPAD_; t += 256) {
        if (t >= G_) s_gbox[t] = make_float4(0.f, 0.f, 1.f, 1.f);
        s_glbl[t] = (t < G_) ? glbl[b * G_ + t] : -1;
        s_gmin[t] = ~0ull;
    }
    __syncthreads();
    for (int t = tid; t < GPAD_; t += 256) {
        float4 gb = s_gbox[t];
        float cx = (gb.x + gb.z) * 0.5f, cy = (gb.y + gb.w) * 0.5f;
        s_gcx[t] = cx; s_gcy[t] = cy; s_gn[t] = cx * cx + cy * cy;
    }
    __syncthreads();
#else
    for (int t = tid; t < GPAD_; t += 256) {
        float4 gb; int lb;
        if (t < G_) { gb = ((const float4*)gbox)[b * G_ + t]; lb = glbl[b * G_ + t]; }
        else        { gb = make_float4(0.f, 0.f, 1.f, 1.f);  lb = -1; }
        s_gbox[t] = gb;
        float cx = (gb.x + gb.z) * 0.5f, cy = (gb.y + gb.w) * 0.5f;
        s_gcx[t] = cx; s_gcy[t] = cy; s_gn[t] = cx * cx + cy * cy;
        s_glbl[t] = lb;
        s_gmin[t] = ~0ull;
    }
    __syncthreads();
#endif

    const int lane = tid & 31;
    const int wave = tid >> 5;
    const int up   = lane >> 4;    // half-wave selector (K/M split per ISA layout)
    const int l15  = lane & 15;

    for (int c = wave; c < 32; c += 8) {                 // uniform per wave -> EXEC all-1s at WMMA
        const int prow  = pblk * 512 + c * 16;           // pred base within image
        const int pbase = b * P_ + prow;                 // global element base

        // ---- A (distance): [-2pcx, -2pcy, |pc|^2, 1] per row M = lane&15
        float4 myb = ((const float4*)pbox)[pbase + l15];
        float pcx = (myb.x + myb.z) * 0.5f, pcy = (myb.y + myb.w) * 0.5f;
        float pn  = pcx * pcx + pcy * pcy;
        v2f aD; aD.x = up ? pn   : (-2.f * pcx);
                aD.y = up ? 1.0f : (-2.f * pcy);

        // ---- A (class): softmax probs of row M, K-split per ISA A layout
        const float4* lg4 = (const float4*)(plog + (size_t)(pbase + l15) * C_);
        float4 lo4 = lg4[0], hi4 = lg4[1];
        float l[8] = { lo4.x, lo4.y, lo4.z, lo4.w, hi4.x, hi4.y, hi4.z, hi4.w };
        float m = l[0];
        #pragma unroll
        for (int k = 1; k < 8; k++) m = fmaxf(m, l[k]);
        float pr[8], s = 0.f;
        #pragma unroll
        for (int k = 0; k < 8; k++) { pr[k] = __expf(l[k] - m); s += pr[k]; }
        float rs = 1.f / s;
        #pragma unroll
        for (int k = 0; k < 8; k++) pr[k] *= rs;
        v2f a2; a2.x = up ? pr[2] : pr[0]; a2.y = up ? pr[3] : pr[1];
        v2f a3; a3.x = up ? pr[6] : pr[4]; a3.y = up ? pr[7] : pr[5];

        // ---- pred boxes for the 8 C-slots this lane owns (rows M = i + 8*up)
        float4 pb[8];
        #pragma unroll
        for (int i = 0; i < 8; i++) pb[i] = ((const float4*)pbox)[pbase + i + up * 8];

        for (int t = 0; t < 10; ++t) {
            const int gcol = t * 16 + l15;
            // B (distance): rows [gcx, gcy, 1, |gc|^2]
            v2f bD; bD.x = up ? 1.0f       : s_gcx[gcol];
                    bD.y = up ? s_gn[gcol] : s_gcy[gcol];
            // B (class): -0.5 * onehot(label), rows 0..3 / 4..7
            int lb = s_glbl[gcol];
            v2f b2, b3;
            b2.x = (lb == (up ? 2 : 0)) ? -0.5f : 0.f;
            b2.y = (lb == (up ? 3 : 1)) ? -0.5f : 0.f;
            b3.x = (lb == (up ? 6 : 4)) ? -0.5f : 0.f;
            b3.y = (lb == (up ? 7 : 5)) ? -0.5f : 0.f;

            v8f accC = {};
            accC = __builtin_amdgcn_wmma_f32_16x16x4_f32(false, a3, false, b3,
                                                         (short)0, accC, false, false);
            accC = __builtin_amdgcn_wmma_f32_16x16x4_f32(false, a2, false, b2,
                                                         (short)0, accC, false, false);
            v8f accD = {};
            accD = __builtin_amdgcn_wmma_f32_16x16x4_f32(false, aD, false, bD,
                                                         (short)0, accD, false, false);

            float4 gb = s_gbox[gcol];
            u64 best = ~0ull;
            #pragma unroll
            for (int i = 0; i < 8; i++) {
                float d2 = fmaxf(accD[i], 1e-12f);
                float nd = __fsqrt_rn(d2) * 0.70710678118654752f;  // /sqrt(2)
                float gi = giou4(pb[i], gb);
                float cost = 2.0f * nd - gi + accC[i];
                u64 pk = ((u64)encf(cost) << 32) | (u32)(prow + i + up * 8);
                best = (pk < best) ? pk : best;
            }
            // combine the two half-waves sharing column N = lane&15
            u64 oth = shflxor64(best, 16);
            best = (oth < best) ? oth : best;
            int gg = t * 16 + lane;
            if (lane < 16 && gg < G_) atomicMin(&s_gmin[gg], best);
        }
    }
    __syncthreads();
    for (int t = tid; t < G_; t += 256)
        atomicMin(&gmin[(size_t)b * GPAD_ + t], s_gmin[t]);
}

// ---------------- kernel 2: validity + conflict-resolution scatter ----------------
__global__ void det_assign(const int* __restrict__ nobj,
                           const u64* __restrict__ gmin,
                           u64* __restrict__ pslot)
{
    int idx = blockIdx.x * blockDim.x + threadIdx.x;
    if (idx >= B_ * G_) return;
    int b = idx / G_, g = idx % G_;
    u64 s = gmin[(size_t)b * GPAD_ + g];
    u32 hi = (u32)(s >> 32);
    u32 assigned = (u32)s;              // pred index (fits low bits)
    if (g < nobj[b] && hi < THRESH_ENC) {
        u64 pk = ((u64)hi << 32) | (u32)g;
        atomicMin(&pslot[(size_t)b * P_ + assigned], pk);
    }
}

// ---------------- kernel 3: per-pred losses, block-reduced ----------------
__global__ __launch_bounds__(256) void det_losses(
    const float* __restrict__ obj,  const float* __restrict__ pbox,
    const float* __restrict__ plog, const float* __restrict__ gbox,
    const int* __restrict__ glbl,   const u64* __restrict__ pslot,
    float* __restrict__ accum)
{
    __shared__ float s_acc[4];
    if (threadIdx.x < 4) s_acc[threadIdx.x] = 0.f;
    __syncthreads();

    int idx = blockIdx.x * blockDim.x + threadIdx.x;   // grid == B_*P_ exactly
    int b = idx / P_;

    __builtin_prefetch(plog + (size_t)idx * C_, 0, 3);  // global_prefetch_b8

    float po = obj[idx];
    po = fminf(fmaxf(po, 1e-7f), 1.f - 1e-7f);
    u64 s = pslot[idx];
    bool matched = (s != ~0ull);
    float bce = matched ? -__logf(po) : -__logf(1.f - po);
    float pt  = matched ? po : (1.f - po);
    float focal = 0.25f * (1.f - pt) * (1.f - pt) * bce;
    atomicAdd(&s_acc[0], focal);

    if (matched) {
        int g = (int)(u32)s;
        float4 mb  = ((const float4*)gbox)[b * G_ + g];
        float4 pbx = ((const float4*)pbox)[idx];
        float gi = giou4(pbx, mb);
        float pa = (pbx.z - pbx.x) / (pbx.w - pbx.y + 1e-6f);
        float ga = (mb.z - mb.x) / (mb.w - mb.y + 1e-6f);
        float da = fabsf(pa - ga);
        float sl1 = (da < 1.f) ? (0.5f * da * da) : (da - 0.5f);
        float bb = (1.f - gi) + sl1;                  // W_ASPECT = 1
        // cross-entropy via log-softmax
        const float4* lg4 = (const float4*)(plog + (size_t)idx * C_);
        float4 lo4 = lg4[0], hi4 = lg4[1];
        float l[8] = { lo4.x, lo4.y, lo4.z, lo4.w, hi4.x, hi4.y, hi4.z, hi4.w };
        float m = l[0];
        #pragma unroll
        for (int k = 1; k < 8; k++) m = fmaxf(m, l[k]);
        float se = 0.f;
        #pragma unroll
        for (int k = 0; k < 8; k++) se += __expf(l[k] - m);
        int lb = glbl[b * G_ + g];
        float ce = (m + __logf(se)) - l[lb];
        atomicAdd(&s_acc[1], bb);
        atomicAdd(&s_acc[2], ce);
        atomicAdd(&s_acc[3], 1.f);
    }
    __syncthreads();
    if (threadIdx.x < 4) atomicAdd(&accum[b * 4 + threadIdx.x], s_acc[threadIdx.x]);
}

// ---------------- kernel 4: finalize (1 wave32) ----------------
__global__ void det_finalize(const float* __restrict__ accum, float* __restrict__ out) {
    int b = threadIdx.x;                // 32 threads == B_
    float o = 0.f, bb = 0.f, cl = 0.f, nm = 0.f;
    if (b < B_) {
        float fs  = accum[b * 4 + 0];
        float bbs = accum[b * 4 + 1];
        float ces = accum[b * 4 + 2];
        float nmb = accum[b * 4 + 3];
        float den = fmaxf(nmb, 1.f);
        o  = fs / (float)P_;
        bb = bbs / den;
        cl = ces / den;
        nm = nmb;
    }
    #pragma unroll
    for (int off = 16; off > 0; off >>= 1) {
        o  += __shfl_down(o,  off, 32);
        bb += __shfl_down(bb, off, 32);
        cl += __shfl_down(cl, off, 32);
        nm += __shfl_down(nm, off, 32);
    }
    if (b == 0) {
        float den  = fmaxf(nm, 1.f);
        float objl = o / (float)B_;
        float bbox = bb / den;
        float cls  = cl / den;
        out[0] = objl + bbox + cls;
        out[1] = objl;
        out[2] = bbox;
        out[3] = cls;
    }
}

// ---------------- launch ----------------
extern "C" void kernel_launch(void* const* d_in, const int* in_sizes, int n_in,
                              void* d_out, int out_size, void* d_ws, size_t ws_size,
                              hipStream_t stream) {
    (void)in_sizes; (void)n_in; (void)out_size; (void)ws_size;
    const float* obj  = (const float*)d_in[0];
    const float* pbox = (const float*)d_in[1];
    const float* plog = (const float*)d_in[2];
    const float* gbox = (const float*)d_in[3];
    const int*   glbl = (const int*)d_in[4];
    const int*   nobj = (const int*)d_in[5];
    float* out = (float*)d_out;

    // workspace layout: [gmin B*GPAD u64][pslot B*P u64][accum B*4 f32]  (~2.1 MB)
    u64*   gmin  = (u64*)d_ws;
    u64*   pslot = gmin + (size_t)B_ * GPAD_;
    float* accum = (float*)(pslot + (size_t)B_ * P_);

    const int nslots = B_ * GPAD_ + B_ * P_;
    const int naccum = B_ * 4;

    det_init_ws<<<(nslots + 255) / 256, 256, 0, stream>>>(gmin, nslots, accum, naccum);
    det_cost_min<<<B_ * PBLOCKS_, 256, 0, stream>>>(pbox, plog, gbox, glbl, gmin);
    det_assign<<<(B_ * G_ + 255) / 256, 256, 0, stream>>>(nobj, gmin, pslot);
    det_losses<<<(B_ * P_) / 256, 256, 0, stream>>>(obj, pbox, plog, gbox, glbl, pslot, accum);
    det_finalize<<<1, 32, 0, stream>>>(accum, out);
}